// DualSwitch_SwapOnly_21973052686634
// MI455X (gfx1250) — compile-verified
//
#include <hip/hip_runtime.h>
#include <hip/hip_bf16.h>

// out[b,c,i,j] = x[b,c, i^3, j^3] for shape (16, 96, 256, 256) fp32.
//
// Pure streaming permutation: j^3 reverses dwords inside each aligned float4,
// i^3 reverses rows inside each aligned group of 4 rows. So each thread moves
// one 16-byte group with a b128 load, a register swizzle, and a b128 store.
// Memory-bound: ~805 MB total traffic -> ~34.6 us floor @ 23.3 TB/s.
// Non-temporal hints: footprint (805 MB) >> L2 (192 MB), every byte used once.

typedef float v4f __attribute__((ext_vector_type(4)));

// Shape constants from the reference (16, 96, 256, 256):
//   W/4 = 64 groups per row, H = 256 rows, 64*256 = 16384 groups per (b,c).
#define GROUPS_PER_ROW_LOG2   6   // 64
#define ROWS_LOG2             8   // 256
#define GROUPS_PER_PLANE_LOG2 14  // 64 * 256

__global__ __launch_bounds__(256) void permute_xor3_kernel(
    const float* __restrict__ x, float* __restrict__ out, unsigned int ngroups) {
  unsigned int g = blockIdx.x * blockDim.x + threadIdx.x;
  if (g >= ngroups) return;

  // Decompose destination group index (all power-of-two bit ops).
  unsigned int j4 = g & ((1u << GROUPS_PER_ROW_LOG2) - 1u);        // group in row
  unsigned int i  = (g >> GROUPS_PER_ROW_LOG2) & ((1u << ROWS_LOG2) - 1u);
  unsigned int bc = g >> GROUPS_PER_PLANE_LOG2;

  // Source group: row i^3 (reversal within aligned 4-row blocks), same j4.
  unsigned int src = (bc << GROUPS_PER_PLANE_LOG2) |
                     ((i ^ 3u) << GROUPS_PER_ROW_LOG2) | j4;

  const v4f* __restrict__ xs = (const v4f*)x;
  v4f*       __restrict__ os = (v4f*)out;

  // Streaming (non-temporal) b128 load, dword reversal (j^3), NT b128 store.
  v4f v = __builtin_nontemporal_load(xs + (size_t)src);
  v4f r = __builtin_shufflevector(v, v, 3, 2, 1, 0);
  __builtin_nontemporal_store(r, os + (size_t)g);
}

extern "C" void kernel_launch(void* const* d_in, const int* in_sizes, int n_in,
                              void* d_out, int out_size, void* d_ws, size_t ws_size,
                              hipStream_t stream) {
  const float* x = (const float*)d_in[0];
  float* out = (float*)d_out;

  // out_size = 16*96*256*256 = 100,663,296 elements -> 25,165,824 float4 groups.
  unsigned int ngroups = (unsigned int)(out_size / 4);
  unsigned int threads = 256;
  unsigned int blocks = (ngroups + threads - 1) / threads;

  permute_xor3_kernel<<<blocks, threads, 0, stream>>>(x, out, ngroups);
}